// Model_79534204388120
// MI455X (gfx1250) — compile-verified
//
#include <hip/hip_runtime.h>
#include <math.h>

#define CH   512
#define TT   2048
#define BB   16
#define KS   5
#define NTOT (BB * TT)      // 32768
#define KDIM (CH * 3)       // 1536
#define MOUT (CH / 2)       // 256
#define KC   128            // K-chunk staged in LDS per iteration

typedef __attribute__((ext_vector_type(2))) float v2f;
typedef __attribute__((ext_vector_type(8))) float v8f;

// ---------------------------------------------------------------------------
// Kernel 1: offset (first 5 channels of the 2K are the only ones consumed) +
// modulator convs (k=3, pad=1); emit per-(b,t,k) sample coefficients:
// wf*mod, wc*mod, floor idx, ceil idx. Branch-free padding (clamp + mask).
// ---------------------------------------------------------------------------
__global__ __launch_bounds__(256)
void offmod_kernel(const float* __restrict__ x,
                   const float* __restrict__ offset_w, const float* __restrict__ offset_b,
                   const float* __restrict__ mod_w,    const float* __restrict__ mod_b,
                   float* __restrict__ wfm, float* __restrict__ wcm,
                   int* __restrict__ pfo, int* __restrict__ pco)
{
    int gid = blockIdx.x * 256 + threadIdx.x;   // over B*T, 256 t per block (one b)
    int b = gid / TT;
    int t = gid % TT;
    const float* xb = x + (size_t)b * CH * TT;

    int   tm1 = max(t - 1, 0),          tp1 = min(t + 1, TT - 1);
    float ml  = (t >= 1) ? 1.f : 0.f,   mr  = (t < TT - 1) ? 1.f : 0.f;

    float aoff[KS] = {0.f, 0.f, 0.f, 0.f, 0.f};
    float amod[KS] = {0.f, 0.f, 0.f, 0.f, 0.f};

    for (int c = 0; c < CH; ++c) {
        const float* xr = xb + (size_t)c * TT;
        float x0 = xr[tm1] * ml;
        float x1 = xr[t];
        float x2 = xr[tp1] * mr;
        const float* ow = offset_w + c * 3;      // [k][c][j], k-stride = KDIM
        const float* mw = mod_w    + c * 3;
#pragma unroll
        for (int k = 0; k < KS; ++k) {
            const float* o = ow + k * KDIM;
            aoff[k] = fmaf(x0, o[0], fmaf(x1, o[1], fmaf(x2, o[2], aoff[k])));
            const float* m = mw + k * KDIM;
            amod[k] = fmaf(x0, m[0], fmaf(x1, m[1], fmaf(x2, m[2], amod[k])));
        }
    }

#pragma unroll
    for (int k = 0; k < KS; ++k) {
        float off = aoff[k] + offset_b[k];
        float md  = 1.f / (1.f + expf(-(amod[k] + mod_b[k])));
        // reference quirk: pos clipped FIRST; weights from clipped floor/ceil
        float pos = (float)t + (float)(k - KS / 2) + off;
        pos = fminf(fmaxf(pos, 0.f), (float)(TT - 1));
        int pf = (int)floorf(pos); pf = min(max(pf, 0), TT - 1);
        int pc = (int)ceilf(pos);  pc = min(max(pc, 0), TT - 1);
        float wf = (float)pc - pos;
        float wc = pos - (float)pf;
        size_t idx = (size_t)gid * KS + k;
        wfm[idx] = wf * md;
        wcm[idx] = wc * md;
        pfo[idx] = pf;
        pco[idx] = pc;
    }
}

// ---------------------------------------------------------------------------
// Kernel 2: deformable gather-MAC using only diag(weight). Thread per t,
// block per (c,b): gathers x[b,c,pf(t,k)] (near-coalesced, pf ~ t+k-2).
// ---------------------------------------------------------------------------
__global__ __launch_bounds__(256)
void deform_kernel(const float* __restrict__ x,
                   const float* __restrict__ weight,           // (C,C,K)
                   const float* __restrict__ wfm, const float* __restrict__ wcm,
                   const int* __restrict__ pfo, const int* __restrict__ pco,
                   float* __restrict__ deformed)
{
    int t = blockIdx.x * 256 + threadIdx.x;
    int c = blockIdx.y;
    int b = blockIdx.z;
    const float* xr = x + ((size_t)b * CH + c) * TT;

    float wd[KS];
#pragma unroll
    for (int k = 0; k < KS; ++k)
        wd[k] = weight[((size_t)c * CH + c) * KS + k];   // diagonal, block-uniform

    size_t base = ((size_t)b * TT + t) * KS;
    float acc = 0.f;
#pragma unroll
    for (int k = 0; k < KS; ++k) {
        float s = fmaf(xr[pfo[base + k]], wfm[base + k],
                       xr[pco[base + k]] * wcm[base + k]);
        acc = fmaf(wd[k], s, acc);
    }
    deformed[((size_t)b * CH + c) * TT + t] = acc;
}

// ---------------------------------------------------------------------------
// Kernel 3: repack rp1_w (256 x [c][j]) into j-major K, K-pair-interleaved,
// M-contiguous layout: Wt2[(kk&~1)*MOUT + 2*m + (kk&1)], kk = j*CH + c.
// A-operand then loads one contiguous b64 per lane per k-step.
// ---------------------------------------------------------------------------
__global__ __launch_bounds__(256)
void wt_kernel(const float* __restrict__ rp1_w, float* __restrict__ Wt2)
{
    int i  = blockIdx.x * 256 + threadIdx.x;    // over MOUT*KDIM
    int m  = i / KDIM;
    int r  = i % KDIM;
    int c  = r / 3;
    int j  = r - 3 * c;
    int kk = j * CH + c;                         // j-major flatten
    Wt2[(size_t)(kk & ~1) * MOUT + 2 * m + (kk & 1)] = rp1_w[i];
}

// ---------------------------------------------------------------------------
// Kernel 4: fused rp1 conv (fp32 WMMA GEMM, M=256 K=1536 N=32768) + bias +
// ReLU + rp2 dot -> strength. Block = 512 thr = 16 waves covering all 16
// M-tiles of one 16-wide N-tile; B panel staged in LDS per 128-K chunk
// (j constant per chunk -> no div/mod, branch-free edge masking).
// h never touches memory.
// ---------------------------------------------------------------------------
__global__ __launch_bounds__(512)
void rp1_wmma_kernel(const float* __restrict__ deformed,
                     const float* __restrict__ Wt2,
                     const float* __restrict__ rp1_b,
                     const float* __restrict__ rp2_w,
                     const float* __restrict__ rp2_b,
                     float* __restrict__ strength)
{
    __shared__ float Bp[KC * 16];    // B panel: 128 K x 16 N
    __shared__ float red[512];       // cross-wave strength reduction

    const int tid  = threadIdx.x;
    const int lane = tid & 31;
    const int wave = tid >> 5;       // 16 waves = 16 M-tiles
    const int half = lane >> 4;      // K-half selector for A/B operands
    const int idx  = lane & 15;      // M index (A) / N index (B,C)

    const int n0    = blockIdx.x * 16;           // N tile; T%16==0 -> one b
    const int bb    = n0 >> 11;                  // n / T
    const int tbase = n0 & (TT - 1);
    const int m0    = wave * 16;
    const int m     = m0 + idx;
    const float* dB = deformed + (size_t)bb * CH * TT;

    // cooperative-load geometry: element e = i*512 + tid
    const int nn  = tid & 15;                    // N within tile
    const int klo = tid >> 4;                    // K_local base (+32 per i)
    const int tn  = tbase + nn;                  // this element's t

    v8f acc = {};
    for (int chunk = 0; chunk < KDIM / KC; ++chunk) {
        const int j     = chunk >> 2;            // 4 chunks per j segment
        const int cbase = chunk * KC - j * CH;   // c = cbase + k_local

        // branch-free padded sample position for this thread's column
        int   tj  = tn + j - 1;
        int   tcl = min(max(tj, 0), TT - 1);
        float msk = (tj >= 0 && tj < TT) ? 1.f : 0.f;

        __syncthreads();                         // previous chunk consumed
#pragma unroll
        for (int i = 0; i < (KC * 16) / 512; ++i) {
            int kl = klo + i * 32;
            Bp[kl * 16 + nn] = dB[(size_t)(cbase + kl) * TT + tcl] * msk;
        }
        __syncthreads();                         // panel ready

        const float* ap = Wt2 + (size_t)(chunk * KC) * MOUT + 2 * m;
#pragma unroll 4
        for (int s = 0; s < KC / 4; ++s) {
            int kbl = s * 4 + 2 * half;          // K_local for this lane-half
            // A: 16x4 fp32; lanes0-15 K={kb,kb+1}, lanes16-31 K={kb+2,kb+3}
            v2f a = *(const v2f*)(ap + (size_t)kbl * MOUT);
            // B: 4x16 fp32 from LDS, N across lanes
            v2f bv;
            bv.x = Bp[kbl * 16 + idx];
            bv.y = Bp[kbl * 16 + 16 + idx];
            // (neg_a, A, neg_b, B, c_mod, C, reuse_a, reuse_b)
            acc = __builtin_amdgcn_wmma_f32_16x16x4_f32(
                false, a, false, bv, (short)0, acc, false, false);
        }
    }

    // epilogue: bias + ReLU + weighted partial sum over this wave's 16 M rows
    // C/D layout: VGPR r -> M = m0 + r + 8*half, N = n0 + idx
    float partial = 0.f;
#pragma unroll
    for (int r = 0; r < 8; ++r) {
        int mm = m0 + r + 8 * half;
        float v = fmaxf(acc[r] + rp1_b[mm], 0.f);
        partial = fmaf(v, rp2_w[mm], partial);
    }
    red[wave * 32 + lane] = partial;
    __syncthreads();

    if (tid < 16) {                              // fixed-order, deterministic
        float s = rp2_b[0];
        for (int w = 0; w < 16; ++w)
            s += red[w * 32 + tid] + red[w * 32 + 16 + tid];
        strength[(size_t)n0 + tid] = s;
    }
}

// ---------------------------------------------------------------------------
extern "C" void kernel_launch(void* const* d_in, const int* in_sizes, int n_in,
                              void* d_out, int out_size, void* d_ws, size_t ws_size,
                              hipStream_t stream)
{
    const float* x        = (const float*)d_in[0];
    const float* offset_w = (const float*)d_in[1];
    const float* offset_b = (const float*)d_in[2];
    const float* mod_w    = (const float*)d_in[3];
    const float* mod_b    = (const float*)d_in[4];
    const float* weight   = (const float*)d_in[5];
    const float* rp1_w    = (const float*)d_in[6];
    const float* rp1_b    = (const float*)d_in[7];
    const float* rp2_w    = (const float*)d_in[8];
    const float* rp2_b    = (const float*)d_in[9];

    float* deformed = (float*)d_out;                        // B*C*T
    float* strength = (float*)d_out + (size_t)BB * CH * TT; // B*T

    // workspace layout (32-bit units): ~4.2 MB total
    float* ws_f = (float*)d_ws;
    const size_t NK = (size_t)BB * TT * KS;    // 163840
    float* wfm = ws_f;
    float* wcm = ws_f + NK;
    int*   pfo = (int*)(ws_f + 2 * NK);
    int*   pco = (int*)(ws_f + 3 * NK);
    float* Wt2 = ws_f + 4 * NK;                // 1536*256

    offmod_kernel<<<dim3(NTOT / 256), dim3(256), 0, stream>>>(
        x, offset_w, offset_b, mod_w, mod_b, wfm, wcm, pfo, pco);

    deform_kernel<<<dim3(TT / 256, CH, BB), dim3(256), 0, stream>>>(
        x, weight, wfm, wcm, pfo, pco, deformed);

    wt_kernel<<<dim3((MOUT * KDIM) / 256), dim3(256), 0, stream>>>(rp1_w, Wt2);

    rp1_wmma_kernel<<<dim3(NTOT / 16), dim3(512), 0, stream>>>(
        deformed, Wt2, rp1_b, rp2_w, rp2_b, strength);
}